// LevinsonDurbin_91225105367394
// MI455X (gfx1250) — compile-verified
//
#include <hip/hip_runtime.h>
#include <cstdint>
#include <cmath>

// Levinson-Durbin for 131072 independent order-32 Toeplitz systems (LPC).
// Thread-per-problem recursion; workload is memory-bound (~35 MB total traffic
// -> ~1.5 us floor at 23.3 TB/s; ~1k VALU/lane keeps compute on par with that).
// CDNA5 paths used:
//   * TDM tensor_load_to_lds (TENSORcnt) — one DMA per wave stages the wave's
//     contiguous 4224 B input tile into LDS (hand-built D# descriptor).
//   * global_store_async_from_lds_b32 (ASYNCcnt) — coalesced LDS->HBM writeback.
//   * wave32-native layout: one problem per lane, stride-33 LDS = conflict-free.

namespace {

constexpr int kOrder         = 32;          // M
constexpr int kVals          = kOrder + 1;  // 33 floats per problem (r and out)
constexpr int kThreads       = 256;         // 8 wave32s per block
constexpr int kWaves         = kThreads / 32;
constexpr int kProbsPerWave  = 32;          // one problem per lane
constexpr int kProbsPerBlock = kWaves * kProbsPerWave;   // 256
constexpr int kDwPerWave     = kProbsPerWave * kVals;    // 1056 dwords / wave

typedef __attribute__((ext_vector_type(4))) unsigned uint4v;
typedef __attribute__((ext_vector_type(8))) unsigned uint8v;

// Low 32 bits of a flat shared-aperture address == LDS byte offset on amdgcn.
__device__ __forceinline__ unsigned lds_off(const void* p) {
    return (unsigned)(uintptr_t)p;
}

__global__ __launch_bounds__(kThreads)
void levinson_kernel(const float* __restrict__ r, float* __restrict__ out,
                     int nprob) {
    __shared__ float smem[kWaves * kDwPerWave];   // 33792 B of the 320 KB LDS
    const int lane = threadIdx.x & 31;
    const int wave = threadIdx.x >> 5;
    const int p0   = blockIdx.x * kProbsPerBlock + wave * kProbsPerWave;
    if (p0 >= nprob) return;   // wave-uniform guard (nprob is a multiple of 32)

    float* wbase = &smem[wave * kDwPerWave];

    // ---- Stage in: one TDM descriptor per wave, global -> LDS -------------
    // D# per CDNA5 ISA: 1D tile of 1056 dwords.
    {
        const unsigned wave_s = __builtin_amdgcn_readfirstlane((unsigned)wave);
        const unsigned p0_s   = (unsigned)blockIdx.x * kProbsPerBlock +
                                wave_s * kProbsPerWave;
        const uint64_t gaddr  = (uint64_t)(uintptr_t)r +
                                (uint64_t)p0_s * (kVals * 4u);
        const unsigned ldsb   = __builtin_amdgcn_readfirstlane(lds_off(wbase));

        const uint4v g0 = {
            1u,                                               // count=1 (valid), no gather
            ldsb,                                             // lds_addr (bytes)
            (unsigned)gaddr,                                  // global_addr[31:0]
            (unsigned)((gaddr >> 32) & 0x01FFFFFFull)         // global_addr[56:32]
                | (2u << 30)                                  // type=2 ("image")
        };
        const uint8v g1 = {
            (2u << 16),                                       // wg_mask=0, data_size=4B
            ((unsigned)kDwPerWave & 0xFFFFu) << 16,           // tensor_dim0[15:0] @bit48
            (((unsigned)kDwPerWave >> 16) & 0xFFFFu)          // tensor_dim0[31:16]
                | (1u << 16),                                 // tensor_dim1=1 @bit80
            0u | ((unsigned)kDwPerWave << 16),                // tile_dim0=1056 @bit112
            1u,                                               // tile_dim1=1, tile_dim2=0
            (unsigned)kDwPerWave,                             // tensor_dim0_stride[31:0]
            0u,                                               // stride0 hi / stride1 lo
            0u                                                // tensor_dim1_stride hi
        };
        asm volatile("tensor_load_to_lds %0, %1" :: "s"(g0), "s"(g1) : "memory");
    }
    asm volatile("s_wait_tensorcnt 0x0" ::: "memory");

    // Each lane reads its own record; stride 33 dwords is odd -> conflict-free.
    float rr[kVals];
    #pragma unroll
    for (int e = 0; e < kVals; ++e) rr[e] = wbase[lane * kVals + e];

    // ---- Levinson-Durbin recursion (fully unrolled, all in VGPRs) ---------
    // Solves  r[m] + sum_{i=1..M} a[i]*r[|m-i|] = 0, m=1..M  (i.e. R a = -r1),
    // with E_M = r0 * prod(1 - k_m^2) = r0 + sum a[i] r[i] = K^2.
    float a[kOrder + 1];
    float E = rr[0];
    #pragma unroll
    for (int m = 1; m <= kOrder; ++m) {
        // Dot product with two independent accumulators (halved dep-chain).
        float acc0 = rr[m], acc1 = 0.0f;
        #pragma unroll
        for (int i = 1; i + 1 < m; i += 2) {
            acc0 = fmaf(a[i],     rr[m - i],     acc0);
            acc1 = fmaf(a[i + 1], rr[m - i - 1], acc1);
        }
        if (((m - 1) & 1) != 0)                 // leftover odd element
            acc0 = fmaf(a[m - 1], rr[1], acc0);
        const float k = -(acc0 + acc1) * __builtin_amdgcn_rcpf(E);

        #pragma unroll
        for (int i = 1; i <= (m - 1) / 2; ++i) {
            const float ai = a[i], aj = a[m - i];
            a[i]     = fmaf(k, aj, ai);
            a[m - i] = fmaf(k, ai, aj);
        }
        if ((m & 1) == 0) {                     // self-paired middle element
            const float ah = a[m / 2];
            a[m / 2] = fmaf(k, ah, ah);
        }
        a[m] = k;
        E = E * (1.0f - k * k);
    }

    // ---- Stage out through LDS, then async LDS -> global ------------------
    wbase[lane * kVals + 0] = sqrtf(E);         // K
    #pragma unroll
    for (int i = 1; i <= kOrder; ++i) wbase[lane * kVals + i] = a[i];
    asm volatile("s_wait_dscnt 0x0" ::: "memory");  // DS writes visible to async engine

    float* gdst = out + (size_t)p0 * kVals;
    #pragma unroll
    for (int j = 0; j < kVals; ++j) {
        float*   g = gdst + j * 32 + lane;
        unsigned l = lds_off(&wbase[j * 32 + lane]);
        asm volatile("global_store_async_from_lds_b32 %0, %1, off"
                     :: "v"(g), "v"(l) : "memory");
    }
    // S_ENDPGM performs an implicit wait-idle, covering outstanding async stores.
}

} // namespace

extern "C" void kernel_launch(void* const* d_in, const int* in_sizes, int n_in,
                              void* d_out, int out_size, void* d_ws, size_t ws_size,
                              hipStream_t stream) {
    (void)n_in; (void)out_size; (void)d_ws; (void)ws_size;
    const float* r   = (const float*)d_in[0];
    float*       out = (float*)d_out;
    const int nprob  = in_sizes[0] / kVals;  // 32*4096 = 131072 for reference shapes
    const int blocks = (nprob + kProbsPerBlock - 1) / kProbsPerBlock;
    levinson_kernel<<<blocks, kThreads, 0, stream>>>(r, out, nprob);
}